// MyGAE_64982855189201
// MI455X (gfx1250) — compile-verified
//
#include <hip/hip_runtime.h>

#define N_NODES 100000
#define NTILES  (N_NODES / 16)   // 6250, exact

typedef __attribute__((ext_vector_type(2))) float v2f;
typedef __attribute__((ext_vector_type(8))) float v8f;

// ---------------- init: deg=1.0 (self loop), zero agg1 and d_out ----------------
__global__ __launch_bounds__(256) void gcn_init(float* deg, float* agg1, float* out,
                                                int nOut, int nAgg, int nDeg) {
    int i = blockIdx.x * 256 + threadIdx.x;
    if (i < nOut) out[i] = 0.0f;
    if (i < nAgg) agg1[i] = 0.0f;
    if (i < nDeg) deg[i] = 1.0f;
}

// ---------------- degree accumulation: deg[dst] += ew ----------------
__global__ __launch_bounds__(256) void gcn_deg(const int* __restrict__ dst,
                                               const float* __restrict__ ew,
                                               float* deg, int E) {
    int e = blockIdx.x * 256 + threadIdx.x;
    if (e < E) atomicAdd(&deg[dst[e]], ew[e]);
}

// ---------------- dinv = rsqrt(deg) in place ----------------
__global__ __launch_bounds__(256) void gcn_dinv(float* deg, int n) {
    int i = blockIdx.x * 256 + threadIdx.x;
    if (i < n) {
        float d = deg[i];
        deg[i] = d > 0.0f ? rsqrtf(d) : 0.0f;
    }
}

// ---------------- GEMM1: xw[N,16] = x[N,128] @ W1[128,16] via V_WMMA_F32_16X16X4_F32 ----------------
__global__ __launch_bounds__(256) void gcn_gemm1(const float* __restrict__ x,
                                                 const float* __restrict__ W1,
                                                 float* __restrict__ xw, int ntiles) {
    int lane = threadIdx.x & 31;
    int tile = blockIdx.x * 8 + (threadIdx.x >> 5);
    if (tile >= ntiles) return;              // wave-uniform: EXEC stays all-1s
    int m  = lane & 15;                      // row (A) / col (B,C)
    int kh = (lane >> 4) << 1;               // lanes 16-31 hold K+2
    const float* xr = x + (size_t)(tile * 16 + m) * 128;
    v8f c = {};
    #pragma unroll
    for (int k0 = 0; k0 < 128; k0 += 4) {
        int k = k0 + kh;
        v2f a; a.x = xr[k];            a.y = xr[k + 1];
        v2f b; b.x = W1[k * 16 + m];   b.y = W1[(k + 1) * 16 + m];
        c = __builtin_amdgcn_wmma_f32_16x16x4_f32(false, a, false, b,
                                                  (short)0, c, false, false);
    }
    int rowbase = tile * 16 + ((lane >> 4) << 3);
    #pragma unroll
    for (int r = 0; r < 8; r++)
        xw[(size_t)(rowbase + r) * 16 + m] = c[r];
}

// ---------------- edge scatter layer 1: 4 threads/edge, 4 floats each ----------------
__global__ __launch_bounds__(256) void gcn_edge1(const int* __restrict__ src,
                                                 const int* __restrict__ dst,
                                                 const float* __restrict__ ew,
                                                 const float* __restrict__ dinv,
                                                 const float* __restrict__ xw,
                                                 float* agg1, int E) {
    int t = blockIdx.x * 256 + threadIdx.x;
    if (t >= E * 4) return;
    int e = t >> 2, q = t & 3;
    int s = src[e], d = dst[e];
    float c = dinv[s] * ew[e] * dinv[d];
    float4 v = ((const float4*)(xw + (size_t)s * 16))[q];
    float* o = agg1 + (size_t)d * 16 + q * 4;
    atomicAdd(o + 0, v.x * c);
    atomicAdd(o + 1, v.y * c);
    atomicAdd(o + 2, v.z * c);
    atomicAdd(o + 3, v.w * c);
}

// ---------------- h = relu(agg1 + xw * dinv^2 + b1), in place over agg1 ----------------
__global__ __launch_bounds__(256) void gcn_h(const float* __restrict__ xw,
                                             const float* __restrict__ dinv,
                                             const float* __restrict__ b1,
                                             float* h_io, int total) {
    int i = blockIdx.x * 256 + threadIdx.x;
    if (i >= total) return;
    int node = i >> 4, col = i & 15;
    float di = dinv[node];
    float v = h_io[i] + xw[i] * di * di + b1[col];
    h_io[i] = v > 0.0f ? v : 0.0f;
}

// ---------------- GEMM2: xw2[N,32] = h[N,16] @ W2[16,32], two N-tiles per wave ----------------
__global__ __launch_bounds__(256) void gcn_gemm2(const float* __restrict__ h,
                                                 const float* __restrict__ W2,
                                                 float* __restrict__ xw2, int ntiles) {
    int lane = threadIdx.x & 31;
    int tile = blockIdx.x * 8 + (threadIdx.x >> 5);
    if (tile >= ntiles) return;
    int m  = lane & 15;
    int kh = (lane >> 4) << 1;
    const float* hr = h + (size_t)(tile * 16 + m) * 16;
    v8f c0 = {}, c1 = {};
    #pragma unroll
    for (int k0 = 0; k0 < 16; k0 += 4) {
        int k = k0 + kh;
        v2f a;  a.x  = hr[k];                 a.y  = hr[k + 1];
        v2f b0; b0.x = W2[k * 32 + m];        b0.y = W2[(k + 1) * 32 + m];
        v2f b1; b1.x = W2[k * 32 + 16 + m];   b1.y = W2[(k + 1) * 32 + 16 + m];
        c0 = __builtin_amdgcn_wmma_f32_16x16x4_f32(false, a, false, b0,
                                                   (short)0, c0, false, false);
        c1 = __builtin_amdgcn_wmma_f32_16x16x4_f32(false, a, false, b1,
                                                   (short)0, c1, false, false);
    }
    int rowbase = tile * 16 + ((lane >> 4) << 3);
    #pragma unroll
    for (int r = 0; r < 8; r++) {
        xw2[(size_t)(rowbase + r) * 32 + m]      = c0[r];
        xw2[(size_t)(rowbase + r) * 32 + 16 + m] = c1[r];
    }
}

// ---------------- edge scatter layer 2: 8 threads/edge, 4 floats each, into d_out ----------------
__global__ __launch_bounds__(256) void gcn_edge2(const int* __restrict__ src,
                                                 const int* __restrict__ dst,
                                                 const float* __restrict__ ew,
                                                 const float* __restrict__ dinv,
                                                 const float* __restrict__ xw2,
                                                 float* out, int E) {
    int t = blockIdx.x * 256 + threadIdx.x;
    if (t >= E * 8) return;
    int e = t >> 3, q = t & 7;
    int s = src[e], d = dst[e];
    float c = dinv[s] * ew[e] * dinv[d];
    float4 v = ((const float4*)(xw2 + (size_t)s * 32))[q];
    float* o = out + (size_t)d * 32 + q * 4;
    atomicAdd(o + 0, v.x * c);
    atomicAdd(o + 1, v.y * c);
    atomicAdd(o + 2, v.z * c);
    atomicAdd(o + 3, v.w * c);
}

// ---------------- out = out(agg) + xw2 * dinv^2 + b2 ----------------
__global__ __launch_bounds__(256) void gcn_out(const float* __restrict__ xw2,
                                               const float* __restrict__ dinv,
                                               const float* __restrict__ b2,
                                               float* out_io, int total) {
    int i = blockIdx.x * 256 + threadIdx.x;
    if (i >= total) return;
    int node = i >> 5, col = i & 31;
    float di = dinv[node];
    out_io[i] = out_io[i] + xw2[i] * di * di + b2[col];
}

extern "C" void kernel_launch(void* const* d_in, const int* in_sizes, int n_in,
                              void* d_out, int out_size, void* d_ws, size_t ws_size,
                              hipStream_t stream) {
    const float* x  = (const float*)d_in[0];
    const int*   ei = (const int*)d_in[1];
    const float* ew = (const float*)d_in[2];
    const float* W1 = (const float*)d_in[3];
    const float* b1 = (const float*)d_in[4];
    const float* W2 = (const float*)d_in[5];
    const float* b2 = (const float*)d_in[6];
    float* out = (float*)d_out;

    const int E = in_sizes[2];          // 3,200,000
    const int* src = ei;
    const int* dst = ei + E;

    // workspace layout (256B aligned)
    char* ws = (char*)d_ws;
    const size_t OFF_DINV = 0;
    const size_t OFF_XW   = ((size_t)N_NODES * 4 + 255) / 256 * 256;        // 400128
    const size_t OFF_H    = OFF_XW + (size_t)N_NODES * 16 * 4;              // +6.4MB
    const size_t OFF_XW2  = OFF_H  + (size_t)N_NODES * 16 * 4;              // +6.4MB
    float* dinv = (float*)(ws + OFF_DINV);  // [N]
    float* xw   = (float*)(ws + OFF_XW);    // [N,16]
    float* h    = (float*)(ws + OFF_H);     // [N,16] : agg1 then h (in place)
    float* xw2  = (float*)(ws + OFF_XW2);   // [N,32]

    const int nH   = N_NODES * 16;          // 1.6M
    const int nOut = N_NODES * 32;          // 3.2M

    gcn_init <<<(nOut + 255) / 256, 256, 0, stream>>>(dinv, h, out, nOut, nH, N_NODES);
    gcn_deg  <<<(E + 255) / 256,    256, 0, stream>>>(dst, ew, dinv, E);
    gcn_dinv <<<(N_NODES + 255) / 256, 256, 0, stream>>>(dinv, N_NODES);

    gcn_gemm1<<<(NTILES + 7) / 8, 256, 0, stream>>>(x, W1, xw, NTILES);
    gcn_edge1<<<(E * 4 + 255) / 256, 256, 0, stream>>>(src, dst, ew, dinv, xw, h, E);
    gcn_h    <<<(nH + 255) / 256, 256, 0, stream>>>(xw, dinv, b1, h, nH);

    gcn_gemm2<<<(NTILES + 7) / 8, 256, 0, stream>>>(h, W2, xw2, NTILES);
    gcn_edge2<<<(E * 8 + 255) / 256, 256, 0, stream>>>(src, dst, ew, dinv, xw2, out, E);
    gcn_out  <<<(nOut + 255) / 256, 256, 0, stream>>>(xw2, dinv, b2, out, nOut);
}